// getTargets_91130616087109
// MI455X (gfx1250) — compile-verified
//
#include <hip/hip_runtime.h>
#include <stdint.h>

// ---------------------------------------------------------------------------
// SimOTA-style target assignment for MI455X (gfx1250, wave32).
// Memory-bound (~190MB HBM traffic -> ~8us floor at 23.3 TB/s); the real cost
// driver is phase-1 block latency, so the top-k threshold now uses a 4-pass
// 256-bin radix select (exact order statistic) instead of a 31-pass bisection.
// gfx1250-specific paths: async global->LDS copy (+ s_wait_asynccnt), wave32
// ballot list compaction, non-temporal b64 output stores.
// ---------------------------------------------------------------------------

namespace cfg {
constexpr int   BSZ    = 16;
constexpr int   H      = 384;
constexpr int   W      = 640;
constexpr int   NBOX   = 32;
constexpr int   NCLS   = 2;
constexpr int   PAIRF  = 16;      // floats per box descriptor
constexpr int   TILE_W = 32;
constexpr int   TILE_H = 8;
constexpr int   STAGE  = 8192;    // LDS IoU staging capacity (rect max ~5084)
constexpr float MODEL_W = 1280.0f;
constexpr float MODEL_H = 768.0f;
constexpr float SCALE   = 80.0f;
}

typedef float f32x2 __attribute__((ext_vector_type(2)));

__device__ __forceinline__ void nt_store2(float* p, float a, float b)
{
    f32x2 v;
    v.x = a; v.y = b;
    __builtin_nontemporal_store(v, (f32x2*)p);   // global_store_b64 th:NT
}

__device__ __forceinline__ float iou_at(const float* __restrict__ pb, int b,
                                        int ih, int iw,
                                        float bminx, float bminy,
                                        float bmaxx, float bmaxy, float barea)
{
    using namespace cfg;
    const int hw = H * W;
    const float* base = pb + (size_t)b * 4 * hw + (size_t)ih * W + iw;
    float p0 = base[0];
    float p1 = base[hw];
    float p2 = base[2 * hw];
    float p3 = base[3 * hw];
    const float sx = MODEL_W / (float)W;   // 2.0
    const float sy = MODEL_H / (float)H;   // 2.0
    float rx = (float)iw * sx + sx * 0.5f;
    float ry = (float)ih * sy + sy * 0.5f;
    float x1 = p0 * SCALE + rx, y1 = p1 * SCALE + ry;
    float x2 = p2 * SCALE + rx, y2 = p3 * SCALE + ry;
    float pw = x2 - x1, ph = y2 - y1;
    float cx = x1 + pw * 0.5f, cy = y1 + ph * 0.5f;
    float pminx = cx - pw * 0.5f, pminy = cy - ph * 0.5f;
    float pmaxx = cx + pw * 0.5f, pmaxy = cy + ph * 0.5f;
    float ix = fmaxf(fminf(pmaxx, bmaxx) - fmaxf(pminx, bminx), 0.0f);
    float iy = fmaxf(fminf(pmaxy, bmaxy) - fmaxf(pminy, bminy), 0.0f);
    float inter = ix * iy;
    float uni = pw * ph + barea - inter;
    return inter / fmaxf(uni, 1e-6f);
}

// ---------------------------------------------------------------------------
// Phase 1: one block per (image, box). 256 threads = 8 wave32s.
// sum-IoU pass (stages rect IoUs in LDS) + 4-pass radix order statistic.
// ---------------------------------------------------------------------------
__global__ __launch_bounds__(256)
void pair_stats_kernel(const float* __restrict__ pb,
                       const float* __restrict__ bx,
                       float* __restrict__ pairs)
{
    using namespace cfg;
    __shared__ float sredf[256];
    __shared__ int   shist[256];
    __shared__ int   sbc[2];
    __shared__ float siou[STAGE];

    const int pair = blockIdx.x;
    const int b    = pair / NBOX;
    const int tid  = threadIdx.x;

    const float* bp = bx + (size_t)pair * 6;
    float x1 = bp[0], y1 = bp[1], x2 = bp[2], y2 = bp[3];
    float cidf = bp[4], extra = bp[5];
    float w = x2 - x1, h = y2 - y1;
    float cx = x1 + w * 0.5f, cy = y1 + h * 0.5f;
    bool  valid = (w * h) > 0.0f;

    float miw = floorf(fmaxf((cx - w * 0.5f) * (float)W / MODEL_W - 0.5f, 0.0f));
    float mih = floorf(fmaxf((cy - h * 0.5f) * (float)H / MODEL_H - 0.5f, 0.0f));
    float maw = ceilf (fminf((cx + w * 0.5f) * (float)W / MODEL_W - 0.5f, (float)W - 1.0f));
    float mah = ceilf (fminf((cy + h * 0.5f) * (float)H / MODEL_H - 0.5f, (float)H - 1.0f));

    int iw0 = 0, iw1 = -1, ih0 = 0, ih1 = -1;
    if (valid && maw >= miw && mah >= mih) {   // NaN compares false -> empty
        iw0 = (int)miw; iw1 = (int)fminf(maw, (float)(W - 1));
        ih0 = (int)mih; ih1 = (int)fminf(mah, (float)(H - 1));
    }
    int rw    = iw1 - iw0 + 1;
    int rectN = (iw1 >= iw0 && ih1 >= ih0) ? rw * (ih1 - ih0 + 1) : 0;
    bool staged = rectN <= STAGE;

    float bminx = cx - w * 0.5f, bminy = cy - h * 0.5f;
    float bmaxx = cx + w * 0.5f, bmaxy = cy + h * 0.5f;
    float barea = w * h;

    // ---- pass 1: sum of IoU over rect; stage IoUs in LDS ----
    float lsum = 0.0f;
    for (int idx = tid; idx < rectN; idx += 256) {
        int ih = ih0 + idx / rw;
        int iw = iw0 + idx % rw;
        float v = iou_at(pb, b, ih, iw, bminx, bminy, bmaxx, bmaxy, barea);
        if (staged) siou[idx] = v;
        lsum += v;
    }
    sredf[tid] = lsum;
    __syncthreads();
    for (int s = 128; s > 0; s >>= 1) {
        if (tid < s) sredf[tid] += sredf[tid + s];
        __syncthreads();
    }
    float sum = sredf[0];
    __syncthreads();

    // dynamic k and lambda
    float kf  = ceilf(fmaxf(sum, 1.0f));
    float lam = rsqrtf(kf);
    const int HWn = H * W;
    int kk = (kf >= (float)HWn) ? (HWn - 1) : (int)kf;
    if (kk > HWn - 1) kk = HWn - 1;
    if (kk < 1) kk = 1;

    // ---- exact (kk+1)-th largest of iou_f over the full map via MSB-first
    // radix select. Non-negative floats are monotone as uints. The H*W-rectN
    // zeros outside the rect live in bin 0 while the selected prefix is 0.
    unsigned pref = 0u;
    int remaining = kk + 1;
    const int imp = HWn - rectN;
    if (rectN > 0) {
        for (int shift = 24; shift >= 0; shift -= 8) {
            shist[tid] = 0;
            __syncthreads();
            for (int idx = tid; idx < rectN; idx += 256) {
                float v;
                if (staged) {
                    v = siou[idx];
                } else {
                    int ih = ih0 + idx / rw;
                    int iw = iw0 + idx % rw;
                    v = iou_at(pb, b, ih, iw, bminx, bminy, bmaxx, bmaxy, barea);
                }
                unsigned u = __float_as_uint(v);
                if (shift == 24 || (u >> (shift + 8)) == pref)
                    atomicAdd(&shist[(u >> shift) & 0xFFu], 1);
            }
            __syncthreads();
            if (tid == 0) {
                int h0 = shist[0] + ((pref == 0u) ? imp : 0);
                int cum = 0, sel = 0, rem2 = remaining;
                for (int bin = 255; bin >= 0; --bin) {
                    int cb = (bin == 0) ? h0 : shist[bin];
                    int nc = cum + cb;
                    if (nc >= remaining) { sel = bin; rem2 = remaining - cum; break; }
                    cum = nc;
                }
                sbc[0] = sel;
                sbc[1] = rem2;
            }
            __syncthreads();
            pref = (pref << 8) | (unsigned)sbc[0];
            remaining = sbc[1];
        }
    }
    float thresh = (rectN > 0) ? __uint_as_float(pref) : 0.0f;

    // one-hot of (clip(cid,0,NCLS-2)+1)
    int cid = (int)cidf;
    if (cid < 0) cid = 0;
    if (cid > NCLS - 2) cid = NCLS - 2;
    float oh0 = ((cid + 1) == 0) ? 1.0f : 0.0f;
    float oh1 = ((cid + 1) == 1) ? 1.0f : 0.0f;

    if (tid == 0) {
        float* o = pairs + (size_t)pair * PAIRF;
        o[0]  = cx;    o[1]  = cy;   o[2]  = w;    o[3]  = h;
        o[4]  = extra; o[5]  = lam;  o[6]  = thresh;
        o[7]  = valid ? 1.0f : 0.0f;
        o[8]  = miw;   o[9]  = mih;  o[10] = maw;  o[11] = mah;
        o[12] = oh0;   o[13] = oh1;  o[14] = 0.0f; o[15] = 0.0f;
    }
}

// ---------------------------------------------------------------------------
// Phase 2: tiles of 8x32 pixels; box table staged via async global->LDS;
// wave32 ballot builds the tile-intersecting box list; NT output stores.
// ---------------------------------------------------------------------------
__global__ __launch_bounds__(256)
void assign_kernel(const float* __restrict__ pb,
                   const float* __restrict__ pairs,
                   float* __restrict__ out)
{
    using namespace cfg;
    __shared__ float sbox[NBOX * PAIRF];   // 512 floats = 2 KB
    __shared__ int   slist[NBOX];
    __shared__ int   scount;

    const int tilesX = W / TILE_W;
    const int b      = blockIdx.y;
    const int tileX  = blockIdx.x % tilesX;
    const int tileY  = blockIdx.x / tilesX;
    const int tid    = threadIdx.x;

    // ---- gfx1250 async copy of the 2KB box table into LDS (b64 per lane) ----
    {
        unsigned lds_off = (unsigned)(size_t)(&sbox[tid * 2]);  // low 32b = LDS offset
        unsigned goff    = (unsigned)(tid * 8);
        unsigned long long sbase =
            (unsigned long long)(size_t)(pairs + (size_t)b * NBOX * PAIRF);
        asm volatile("global_load_async_to_lds_b64 %0, %1, %2"
                     :: "v"(lds_off), "v"(goff), "s"(sbase) : "memory");
        asm volatile("s_wait_asynccnt 0" ::: "memory");
    }
    __syncthreads();

    // ---- wave32 ballot compaction of boxes intersecting this tile ----
    if (tid < NBOX) {   // NBOX == 32 == one wave
        const float* q = &sbox[tid * PAIRF];
        float x0f = (float)(tileX * TILE_W);
        float x1f = (float)(tileX * TILE_W + TILE_W - 1);
        float y0f = (float)(tileY * TILE_H);
        float y1f = (float)(tileY * TILE_H + TILE_H - 1);
        bool hit = (q[7] != 0.0f) && (q[10] >= x0f) && (q[8] <= x1f) &&
                   (q[11] >= y0f) && (q[9] <= y1f);
        unsigned mask = (unsigned)__ballot(hit);
        if (hit) slist[__popc(mask & ((1u << tid) - 1u))] = tid;
        if (tid == 0) scount = __popc(mask);
    }
    __syncthreads();

    const int iw = tileX * TILE_W + (tid % TILE_W);
    const int ih = tileY * TILE_H + (tid / TILE_W);
    if (iw >= W || ih >= H) return;

    // decode prediction at this pixel once
    const int hw = H * W;
    const float* base = pb + (size_t)b * 4 * hw + (size_t)ih * W + iw;
    __builtin_prefetch(base, 0, 1);
    __builtin_prefetch(base + 2 * hw, 0, 1);
    float p0 = base[0], p1 = base[hw], p2 = base[2 * hw], p3 = base[3 * hw];
    const float sx = MODEL_W / (float)W, sy = MODEL_H / (float)H;
    float rx  = (float)iw * sx + sx * 0.5f;
    float ry  = (float)ih * sy + sy * 0.5f;
    float px1 = p0 * SCALE + rx, py1 = p1 * SCALE + ry;
    float px2 = p2 * SCALE + rx, py2 = p3 * SCALE + ry;
    float pw  = px2 - px1, ph = py2 - py1;
    float pcx = px1 + pw * 0.5f, pcy = py1 + ph * 0.5f;
    float pminx = pcx - pw * 0.5f, pminy = pcy - ph * 0.5f;
    float pmaxx = pcx + pw * 0.5f, pmaxy = pcy + ph * 0.5f;
    float parea = pw * ph;

    float colf = (float)iw, rowf = (float)ih;
    float cls0 = 1.0f, cls1 = 0.0f;
    float pt0 = 1.0f, pt1 = 1.0f, pt2 = 1.0f, pt3 = 1.0f, pt4 = 1.0f, pt5 = 1.0f;
    float best = 0.0f;

    const int cnt = scount;
    for (int j = 0; j < cnt; ++j) {
        const float* q = &sbox[slist[j] * PAIRF];
        float cx = q[0], cy = q[1], bw = q[2], bh = q[3];
        float extra = q[4], lam = q[5], thresh = q[6];
        float miw = q[8], mih = q[9], maw = q[10], mah = q[11];

        bool rect = (colf >= miw) && (colf <= maw) &&
                    (rowf >= mih) && (rowf <= mah);   // list holds valid boxes only

        float bminx = cx - bw * 0.5f, bminy = cy - bh * 0.5f;
        float bmaxx = cx + bw * 0.5f, bmaxy = cy + bh * 0.5f;
        float ixl = fmaxf(fminf(pmaxx, bmaxx) - fmaxf(pminx, bminx), 0.0f);
        float iyl = fmaxf(fminf(pmaxy, bmaxy) - fmaxf(pminy, bminy), 0.0f);
        float inter = ixl * iyl;
        float uni   = parea + bw * bh - inter;
        float iou   = inter / fmaxf(uni, 1e-6f);

        float iou_f = rect ? iou : 0.0f;
        float iou_k = (iou_f > thresh) ? iou_f : 0.0f;
        bool positive = rect && (iou_k > 0.0f);
        bool nw  = positive && (iou_k > best);
        bool tie = positive && (iou_k == best) && (best > 0.0f);
        bool bg  = rect && !(iou_k > 0.0f);

        if (nw)  { cls0 = q[12]; cls1 = q[13]; }
        if (tie) { cls0 = 0.0f;  cls1 = 0.0f; }
        if (bg)  { cls0 = 0.0f; }
        if (nw)  { pt0 = cx; pt1 = cy; pt2 = bw; pt3 = bh;
                   pt4 = extra; pt5 = lam; best = iou_k; }
    }

    // streaming (non-temporal) packed writes: cls (b64), pts (3x b64)
    size_t pix = (size_t)b * hw + (size_t)ih * W + iw;
    nt_store2(out + pix * 2, cls0, cls1);
    float* pp = out + (size_t)BSZ * hw * 2 + pix * 6;
    nt_store2(pp + 0, pt0, pt1);
    nt_store2(pp + 2, pt2, pt3);
    nt_store2(pp + 4, pt4, pt5);
}

// ---------------------------------------------------------------------------
extern "C" void kernel_launch(void* const* d_in, const int* in_sizes, int n_in,
                              void* d_out, int out_size, void* d_ws, size_t ws_size,
                              hipStream_t stream)
{
    using namespace cfg;
    (void)in_sizes; (void)n_in; (void)out_size; (void)ws_size;

    const float* pred_boxes = (const float*)d_in[1];  // (bs,4,H,W)
    const float* bboxes     = (const float*)d_in[2];  // (bs,N,6)
    float*       pairs      = (float*)d_ws;           // bs*N*16 floats = 32 KB
    float*       out        = (float*)d_out;          // cls (bs,H,W,2) ++ pts (bs,H,W,6)

    pair_stats_kernel<<<BSZ * NBOX, 256, 0, stream>>>(pred_boxes, bboxes, pairs);

    dim3 grid((W / TILE_W) * (H / TILE_H), BSZ);
    assign_kernel<<<grid, 256, 0, stream>>>(pred_boxes, pairs, out);
}